// Attention_80960133530355
// MI455X (gfx1250) — compile-verified
//
#include <hip/hip_runtime.h>
#include <math.h>

typedef __attribute__((ext_vector_type(2))) float v2f;
typedef __attribute__((ext_vector_type(4))) float v4f;
typedef __attribute__((ext_vector_type(8))) float v8f;

// Problem constants: A=64 agents, N=128 features, H=8 hidden.
// One workgroup (256 threads = 8 wave32) per (t,b) pair; 16384 workgroups.
//
// Phase map (all matmuls on the WMMA f32 16x16x4 pipe):
//   1. coalesced f32x4 load of X (64x128) -> LDS; Wq|Wk combined (128x16) -> LDS
//   2. Z(64x16) = X @ Wc via WMMA, split-K across wave pairs, reduce + bias,
//      scatter into Q(64x8) and the "reshape" key layout Kmat(8x64)
//   3. S(64x64) = Q @ Kmat via WMMA (16 tiles, 2 per wave, K=8 -> 2 WMMA each)
//   4. row softmax (one thread per row, reciprocal stashed in pad column)
//   5. coalesced off-diagonal gather/store (64x63)

#define XS_STRIDE 132   // 132 % 64 == 4 -> conflict-free A-frag loads, 16B-aligned rows
#define QS_STRIDE 9
#define KM_STRIDE 72
#define SS_STRIDE 65
#define PS_STRIDE 17

__global__ __launch_bounds__(256) void attn_offdiag_kernel(
    const float* __restrict__ x,    // (TB, 64, 128)
    const float* __restrict__ Wq,   // (128, 8)
    const float* __restrict__ bq,   // (8)
    const float* __restrict__ Wk,   // (128, 8)
    const float* __restrict__ bk,   // (8)
    float* __restrict__ out)        // (TB, 64, 63)
{
    __shared__ float Xs[64 * XS_STRIDE];      // ~33 KB
    __shared__ float Wc[128 * 16];            // 8 KB   [n][c], c<8: Wq, c>=8: Wk
    __shared__ float bc[16];
    __shared__ float Qs[64 * QS_STRIDE];      // ~2.3 KB
    __shared__ float Km[8 * KM_STRIDE];       // ~2.3 KB  reshape-layout key
    __shared__ float Ss[64 * SS_STRIDE];      // ~16.6 KB scores; col 64 = 1/rowsum
    __shared__ float Ps[4 * 16 * PS_STRIDE];  // ~4.3 KB  split-K partial tiles

    const int tid  = threadIdx.x;
    const int wave = tid >> 5;
    const int lane = tid & 31;
    const int half = lane >> 4;        // 0: lanes 0-15, 1: lanes 16-31
    const int l16  = lane & 15;
    const long long blk = blockIdx.x;

    // ---- Phase 1: load X (8192 f32, coalesced float4) and weights ----
    {
        const v4f* src = (const v4f*)(x + blk * (64LL * 128LL));
#pragma unroll
        for (int i = 0; i < 8; ++i) {
            const int g   = tid + i * 256;      // float4 index, 0..2047
            const int row = g >> 5;             // 32 float4 per row
            const int c4  = (g & 31) << 2;
            *(v4f*)&Xs[row * XS_STRIDE + c4] = src[g];
        }
    }
#pragma unroll
    for (int i = 0; i < 8; ++i) {
        const int idx = tid + i * 256;          // 0..2047 over Wc[n][c]
        const int n = idx >> 4;
        const int c = idx & 15;
        Wc[idx] = (c < 8) ? Wq[n * 8 + c] : Wk[n * 8 + (c - 8)];
    }
    if (tid < 16) bc[tid] = (tid < 8) ? bq[tid] : bk[tid - 8];
    __syncthreads();

    // ---- Phase 2: Z = X @ Wc via WMMA, split-K across wave pairs ----
    // wave w: M-tile mi = w&3, K-half kh = w>>2 (k-steps kh*16 .. kh*16+15).
    {
        const int mi = wave & 3;
        const int kh = wave >> 2;
        v8f p0 = {}, p1 = {};
        const float* xrow = &Xs[(mi * 16 + l16) * XS_STRIDE];
#pragma unroll
        for (int kc = 0; kc < 16; kc += 2) {
            {
                const int kb = (kh * 16 + kc) * 4 + half * 2;   // abs K, even
                v2f a, b;
                a.x = xrow[kb];
                a.y = xrow[kb + 1];
                b.x = Wc[kb * 16 + l16];
                b.y = Wc[(kb + 1) * 16 + l16];
                p0 = __builtin_amdgcn_wmma_f32_16x16x4_f32(
                        false, a, false, b, (short)0, p0, false, false);
            }
            {
                const int kb = (kh * 16 + kc + 1) * 4 + half * 2;
                v2f a, b;
                a.x = xrow[kb];
                a.y = xrow[kb + 1];
                b.x = Wc[kb * 16 + l16];
                b.y = Wc[(kb + 1) * 16 + l16];
                p1 = __builtin_amdgcn_wmma_f32_16x16x4_f32(
                        false, a, false, b, (short)0, p1, false, false);
            }
        }
        // upper K-half waves publish partials
        if (kh == 1) {
#pragma unroll
            for (int r = 0; r < 8; ++r)
                Ps[mi * (16 * PS_STRIDE) + (half * 8 + r) * PS_STRIDE + l16] =
                    p0[r] + p1[r];
        }
        __syncthreads();
        // lower K-half waves reduce, add bias, scatter to Q / reshape-Kmat
        if (kh == 0) {
#pragma unroll
            for (int r = 0; r < 8; ++r) {
                const int row = mi * 16 + half * 8 + r;
                const float v = p0[r] + p1[r] + bc[l16] +
                    Ps[mi * (16 * PS_STRIDE) + (half * 8 + r) * PS_STRIDE + l16];
                if (l16 < 8) {
                    Qs[row * QS_STRIDE + l16] = v;
                } else {
                    const int m = row * 8 + (l16 - 8);   // flat (A*H) index
                    Km[(m >> 6) * KM_STRIDE + (m & 63)] = v;
                }
            }
        }
        __syncthreads();
    }

    // ---- Phase 3: S(64x64) = Q(64x8) @ Km(8x64) via WMMA ----
    // 16 tiles of 16x16; 2 per wave; K=8 -> 2 WMMA per tile.
#pragma unroll
    for (int tt = 0; tt < 2; ++tt) {
        const int tile = wave * 2 + tt;      // 0..15
        const int mi = tile >> 2;
        const int ni = tile & 3;
        v8f acc = {};
#pragma unroll
        for (int kc = 0; kc < 2; ++kc) {
            const int kb = kc * 4 + half * 2;
            v2f a, b;
            a.x = Qs[(mi * 16 + l16) * QS_STRIDE + kb];
            a.y = Qs[(mi * 16 + l16) * QS_STRIDE + kb + 1];
            b.x = Km[kb * KM_STRIDE + ni * 16 + l16];
            b.y = Km[(kb + 1) * KM_STRIDE + ni * 16 + l16];
            acc = __builtin_amdgcn_wmma_f32_16x16x4_f32(
                    false, a, false, b, (short)0, acc, false, false);
        }
#pragma unroll
        for (int r = 0; r < 8; ++r)
            Ss[(mi * 16 + half * 8 + r) * SS_STRIDE + ni * 16 + l16] = acc[r];
    }
    __syncthreads();

    // ---- Phase 4: row softmax (scale = 1/sqrt(1) = 1) ----
    if (tid < 64) {
        float* srow = &Ss[tid * SS_STRIDE];
        float mx = srow[0];
        for (int j = 1; j < 64; ++j) mx = fmaxf(mx, srow[j]);
        float sum = 0.f;
        for (int j = 0; j < 64; ++j) {
            const float e = expf(srow[j] - mx);
            srow[j] = e;
            sum += e;
        }
        srow[64] = 1.0f / sum;
    }
    __syncthreads();

    // ---- Phase 5: coalesced off-diagonal gather + store (64*63 f32) ----
    float* ob = out + blk * (64LL * 63LL);
    for (int o = tid; o < 64 * 63; o += 256) {
        const int r = o / 63;
        const int j = o - r * 63;
        const int c = j + (j >= r);
        ob[o] = Ss[r * SS_STRIDE + c] * Ss[r * SS_STRIDE + 64];
    }
}

extern "C" void kernel_launch(void* const* d_in, const int* in_sizes, int n_in,
                              void* d_out, int out_size, void* d_ws, size_t ws_size,
                              hipStream_t stream) {
    const float* x  = (const float*)d_in[0];   // (T,B,64,128)
    const float* Wq = (const float*)d_in[1];   // (128,8)
    const float* bq = (const float*)d_in[2];   // (8)
    const float* Wk = (const float*)d_in[3];   // (128,8)
    const float* bk = (const float*)d_in[4];   // (8)
    float* out = (float*)d_out;                // (T,B,64,63)

    const int TB = in_sizes[0] / (64 * 128);   // 512*32 = 16384 blocks
    attn_offdiag_kernel<<<TB, 256, 0, stream>>>(x, Wq, bq, Wk, bk, out);
}